// TensorizedEmbedding_369367188184
// MI455X (gfx1250) — compile-verified
//
#include <hip/hip_runtime.h>

// ---------------------------------------------------------------------------
// TT-embedding for MI455X (gfx1250, wave32, WMMA).
//
// Strategy: fuse (core0,core1) -> T01[80][16x16] and (core2,core3) ->
// T23[400][16x48] once into workspace, PRE-SWIZZLED into V_WMMA_F32_16X16X4_F32
// register-fragment order. Main kernel: one wave per token, 12 WMMAs
// ([16x16]@[16x48], K split into 4 chunks of 4), coalesced b64 fragment loads,
// direct D-layout stores. Output (100.7 MB) is the HBM roofline (~4.4 us).
// ---------------------------------------------------------------------------

typedef float v2f __attribute__((ext_vector_type(2)));
typedef float v8f __attribute__((ext_vector_type(8)));

#define NT01 (80 * 256)    // 80 combos x 16x16 floats
#define NT23 (400 * 768)   // 400 combos x 16x48 floats

// Fragment-position decode shared by both fuse kernels (ISA 7.12.2, f32 A/B):
//   element (kc, lane, v): K = kc*4 + 2*(lane/16) + v ; row/col = lane%16.

__global__ void __launch_bounds__(256)
fuse01_kernel(const float* __restrict__ c0, const float* __restrict__ c1,
              float* __restrict__ T01) {
    const int d01 = blockIdx.x;          // 0..79  (d0*10 + d1)
    const int t   = threadIdx.x;         // 0..255 -> one A-matrix element
    const int d0 = d01 / 10, d1 = d01 % 10;
    const int kc   = t >> 6;             // K-chunk 0..3
    const int rem  = t & 63;
    const int lane = rem >> 1;
    const int v    = rem & 1;
    const int M  = lane & 15;                          // M = m0*4 + m1
    const int r2 = kc * 4 + ((lane >> 4) << 1) + v;    // global K = r2
    const int m0 = M >> 2, m1 = M & 3;
    // core0: [1,8,4,16]  core1: [16,10,4,16]
    float acc = 0.0f;
#pragma unroll
    for (int r1 = 0; r1 < 16; ++r1) {
        const float a = c0[(d0 * 4 + m0) * 16 + r1];
        const float b = c1[((r1 * 10 + d1) * 4 + m1) * 16 + r2];
        acc = fmaf(a, b, acc);
    }
    T01[d01 * 256 + t] = acc;            // stored in A-fragment order
}

__global__ void __launch_bounds__(768)
fuse23_kernel(const float* __restrict__ c2, const float* __restrict__ c3,
              float* __restrict__ T23) {
    const int d23 = blockIdx.x;          // 0..399 (d2*20 + d3)
    const int t   = threadIdx.x;         // 0..767 -> one B-matrix element
    const int d2 = d23 / 20, d3 = d23 % 20;
    const int tkc  = t >> 6;             // 0..11 = tile*4 + kc
    const int tile = tkc >> 2;           // N-tile 0..2
    const int kc   = tkc & 3;            // K-chunk 0..3
    const int rem  = t & 63;
    const int lane = rem >> 1;
    const int v    = rem & 1;
    const int N  = lane & 15;
    const int r2 = kc * 4 + ((lane >> 4) << 1) + v;    // global K = r2
    const int j  = tile * 16 + N;                      // col = m2*8 + m3
    const int m2 = j >> 3, m3 = j & 7;
    // core2: [16,20,6,16]  core3: [16,20,8,1]
    float acc = 0.0f;
#pragma unroll
    for (int r3 = 0; r3 < 16; ++r3) {
        const float a = c2[((r2 * 20 + d2) * 6 + m2) * 16 + r3];
        const float b = c3[(r3 * 20 + d3) * 8 + m3];
        acc = fmaf(a, b, acc);
    }
    T23[d23 * 768 + t] = acc;            // stored in B-fragment order
}

// One wave per token: out[16x48] = T01[d01] (16x16) @ T23[d23] (16x48).
__global__ void __launch_bounds__(256)
tt_embed_kernel(const int* __restrict__ x, const float* __restrict__ T01,
                const float* __restrict__ T23, float* __restrict__ out,
                int ntok) {
    const int wave  = threadIdx.x >> 5;
    const int lane  = threadIdx.x & 31;
    const int token = blockIdx.x * 8 + wave;
    if (token >= ntok) return;           // wave-uniform: EXEC all-1 for WMMA

    const int idx = __builtin_amdgcn_readfirstlane(x[token]);
    const int d01 = idx / 400;           // d0*10 + d1   (idx < 32000)
    const int d23 = idx - d01 * 400;     // d2*20 + d3

    const float* __restrict__ Af = T01 + d01 * 256 + lane * 2;
    const float* __restrict__ Bf = T23 + d23 * 768 + lane * 2;

    // A fragments: 4 K-chunks, 2 VGPRs each (coalesced b64 loads).
    const v2f a0 = *(const v2f*)(Af + 0 * 64);
    const v2f a1 = *(const v2f*)(Af + 1 * 64);
    const v2f a2 = *(const v2f*)(Af + 2 * 64);
    const v2f a3 = *(const v2f*)(Af + 3 * 64);

#define BFRAG(t, k) (*(const v2f*)(Bf + ((t) * 4 + (k)) * 64))
#define WMMA(a, b, c) \
    __builtin_amdgcn_wmma_f32_16x16x4_f32(false, (a), false, (b), (short)0, (c), false, false)

    v8f c0 = {}, c1 = {}, c2 = {};
    // N-tile 0
    c0 = WMMA(a0, BFRAG(0, 0), c0);
    c0 = WMMA(a1, BFRAG(0, 1), c0);
    c0 = WMMA(a2, BFRAG(0, 2), c0);
    c0 = WMMA(a3, BFRAG(0, 3), c0);
    // N-tile 1
    c1 = WMMA(a0, BFRAG(1, 0), c1);
    c1 = WMMA(a1, BFRAG(1, 1), c1);
    c1 = WMMA(a2, BFRAG(1, 2), c1);
    c1 = WMMA(a3, BFRAG(1, 3), c1);
    // N-tile 2
    c2 = WMMA(a0, BFRAG(2, 0), c2);
    c2 = WMMA(a1, BFRAG(2, 1), c2);
    c2 = WMMA(a2, BFRAG(2, 2), c2);
    c2 = WMMA(a3, BFRAG(2, 3), c2);
#undef BFRAG
#undef WMMA

    // D layout: vgpr v, lane l -> D[M = v + 8*(l/16), N = l%16].
    // out column = M*48 + tile*16 + N  (full 768-float row per token).
    float* __restrict__ orow = out + (size_t)token * 768;
    const int N   = lane & 15;
    const int mhi = (lane >> 4) << 3;    // 0 or 8
#pragma unroll
    for (int v = 0; v < 8; ++v) {
        const int rowoff = (mhi + v) * 48;
        orow[rowoff + 0  + N] = c0[v];   // lanes 0-15 / 16-31: 64B segments
        orow[rowoff + 16 + N] = c1[v];
        orow[rowoff + 32 + N] = c2[v];
    }
}

extern "C" void kernel_launch(void* const* d_in, const int* in_sizes, int n_in,
                              void* d_out, int out_size, void* d_ws, size_t ws_size,
                              hipStream_t stream) {
    const int*   x  = (const int*)d_in[0];
    const float* c0 = (const float*)d_in[1];   // [1,8,4,16]
    const float* c1 = (const float*)d_in[2];   // [16,10,4,16]
    const float* c2 = (const float*)d_in[3];   // [16,20,6,16]
    const float* c3 = (const float*)d_in[4];   // [16,20,8,1]
    float* out = (float*)d_out;

    float* T01 = (float*)d_ws;                 // 80*256  floats (80 KB)
    float* T23 = T01 + NT01;                   // 400*768 floats (1.2 MB, L2-resident)

    const int ntok = in_sizes[0];              // 32768 tokens

    fuse01_kernel<<<80, 256, 0, stream>>>(c0, c1, T01);
    fuse23_kernel<<<400, 768, 0, stream>>>(c2, c3, T23);

    const int blocks = (ntok + 7) / 8;         // 8 waves (tokens) per block
    tt_embed_kernel<<<blocks, 256, 0, stream>>>(x, T01, T23, out, ntok);
}